// CustomCLIP_51110110822842
// MI455X (gfx1250) — compile-verified
//
#include <hip/hip_runtime.h>
#include <hip/hip_bf16.h>
#include <math.h>

// Problem constants (from reference)
constexpr int Bv   = 64;
constexpr int DINv = 150528;
constexpr int Dv   = 512;
constexpr int Cv   = 100;
constexpr int Mv   = 64;
constexpr int NFv  = 256;
constexpr float EPSv = 1e-6f;

// GEMM tiling
constexpr int KSPLIT = 32;
constexpr int KCHUNK = DINv / KSPLIT;   // 4704 = 32*147
constexpr int NIT    = KCHUNK / 32;     // 147 K-steps of 32
constexpr int LDSA   = 36;  // A row stride (floats): 144B = 9*16B -> 16B-aligned chunks, conflict-free b64 fetch
constexpr int LDSB   = 80;  // B row stride (floats): 320B; 2*80 mod 64 = 32 -> disjoint half-wave bank ranges

typedef float v2f __attribute__((ext_vector_type(2)));
typedef float v4f __attribute__((ext_vector_type(4)));
typedef float v8f __attribute__((ext_vector_type(8)));
typedef int   v4i __attribute__((ext_vector_type(4)));

#define AS1 __attribute__((address_space(1)))
#define AS3 __attribute__((address_space(3)))

#if defined(__gfx1250__) && __has_builtin(__builtin_amdgcn_global_load_async_to_lds_b128)
#define HAS_ASYNC 1
#else
#define HAS_ASYNC 0
#endif

__device__ __forceinline__ void copy16_to_lds(const float* g, float* l) {
#if HAS_ASYNC
    __builtin_amdgcn_global_load_async_to_lds_b128(
        (AS1 v4i*)(g), (AS3 v4i*)(l), 0, 0);
#else
    *(v4f*)l = *(const v4f*)g;
#endif
}

template <int N>
__device__ __forceinline__ void wait_async() {
#if HAS_ASYNC
#if __has_builtin(__builtin_amdgcn_s_wait_asynccnt)
    __builtin_amdgcn_s_wait_asynccnt(N);
#else
    asm volatile("s_wait_asynccnt %0" :: "n"(N) : "memory");
#endif
#endif
}

// ---------------------------------------------------------------------------
// Kernel 1: split-K GEMM  partial[ks][b][d] = image[b, kchunk] @ Wenc[kchunk, d]
// Block: 256 threads (8 waves). Block tile: 64(M) x 64(N), K step 32.
// Double-buffered LDS, async global->LDS DMA, WMMA f32 16x16x4.
// ---------------------------------------------------------------------------
__global__ __launch_bounds__(256) void gemm_partial_kernel(
    const float* __restrict__ image, const float* __restrict__ Wenc,
    float* __restrict__ partial)
{
    __shared__ float Abuf[2][64 * LDSA];   // m-major: Abuf[b][m*LDSA + k]
    __shared__ float Bbuf[2][32 * LDSB];   // k-major: Bbuf[b][k*LDSB + n]

    const int tid  = threadIdx.x;
    const int lane = tid & 31;
    const int wave = tid >> 5;
    const int ng   = blockIdx.x;          // 0..7  (N group of 64 cols)
    const int ks   = blockIdx.y;          // 0..31 (K chunk)
    const int nbase = ng * 64;
    const int kbase = ks * KCHUNK;

    const int mtile = wave & 3;
    const int nsub0 = (wave >> 2) * 2;
    const int nsub1 = nsub0 + 1;

    v8f acc0 = {};
    v8f acc1 = {};

    // A tile = 64 rows x 32 floats = 512 x 16B chunks -> 2 chunks/thread
    const int ar = tid >> 3;              // 0..31 (rows ar and ar+32)
    const int ac = (tid & 7) * 4;         // 0..28
    // B tile = 32 rows x 64 floats = 512 x 16B chunks -> 2 chunks/thread
    const int br = tid >> 4;              // 0..15 (rows br and br+16)
    const int bc = (tid & 15) * 4;        // 0..60

    const int halfsel = lane >> 4;        // 0 | 1
    const int koff    = halfsel * 2;      // fp32 WMMA A/B operand K offset
    const int l15     = lane & 15;

    // issue one 32-wide K tile into buffer `buf`
    auto issue_tile = [&](int k0, int buf) {
        const float* gA = image + (size_t)ar * DINv + k0 + ac;
        copy16_to_lds(gA,                     &Abuf[buf][ar * LDSA + ac]);
        copy16_to_lds(gA + (size_t)32 * DINv, &Abuf[buf][(ar + 32) * LDSA + ac]);
        const float* gB = Wenc + (size_t)(k0 + br) * Dv + nbase + bc;
        copy16_to_lds(gB,                     &Bbuf[buf][br * LDSB + bc]);
        copy16_to_lds(gB + (size_t)16 * Dv,   &Bbuf[buf][(br + 16) * LDSB + bc]);
    };

    issue_tile(kbase, 0);

    for (int it = 0; it < NIT; ++it) {
        const int cur = it & 1;
        if (it + 1 < NIT) {
            issue_tile(kbase + (it + 1) * 32, cur ^ 1);
            wait_async<4>();   // first 4 (this tile's) async loads done; in-order completion
        } else {
            wait_async<0>();
        }
        __syncthreads();

        const float* arow  = &Abuf[cur][(mtile * 16 + l15) * LDSA + koff];
        const float* bcol0 = &Bbuf[cur][nsub0 * 16 + l15];
        const float* bcol1 = &Bbuf[cur][nsub1 * 16 + l15];
#pragma unroll
        for (int kk = 0; kk < 32; kk += 4) {
            v2f a = *(const v2f*)(arow + kk);
            v2f b0, b1;
            b0.x = bcol0[(kk + koff) * LDSB];
            b0.y = bcol0[(kk + koff + 1) * LDSB];
            b1.x = bcol1[(kk + koff) * LDSB];
            b1.y = bcol1[(kk + koff + 1) * LDSB];
            acc0 = __builtin_amdgcn_wmma_f32_16x16x4_f32(
                       false, a, false, b0, (short)0, acc0, false, false);
            acc1 = __builtin_amdgcn_wmma_f32_16x16x4_f32(
                       false, a, false, b1, (short)0, acc1, false, false);
        }
        __syncthreads();
    }

    // C/D layout: N = lane&15, M = r + 8*(lane>=16)
    const int mbase2 = mtile * 16 + halfsel * 8;
    float* prow = partial + (size_t)ks * Bv * Dv;
#pragma unroll
    for (int r = 0; r < 8; ++r) {
        const int mrow = mbase2 + r;
        prow[(size_t)mrow * Dv + nbase + nsub0 * 16 + l15] = acc0[r];
        prow[(size_t)mrow * Dv + nbase + nsub1 * 16 + l15] = acc1[r];
    }
}

// ---------------------------------------------------------------------------
// Kernel 2: sum partials over KSPLIT and L2-normalize each row of f
// ---------------------------------------------------------------------------
__global__ __launch_bounds__(256) void reduce_norm_kernel(
    const float* __restrict__ partial, float* __restrict__ f)
{
    const int b   = blockIdx.x;
    const int tid = threadIdx.x;

    float s0 = 0.f, s1 = 0.f;
    for (int ks = 0; ks < KSPLIT; ++ks) {
        const float* p = partial + ((size_t)ks * Bv + b) * Dv;
        s0 += p[tid];
        s1 += p[tid + 256];
    }
    float loc = s0 * s0 + s1 * s1;
#pragma unroll
    for (int off = 16; off > 0; off >>= 1) loc += __shfl_xor(loc, off, 32);

    __shared__ float red[8];
    __shared__ float invn;
    if ((tid & 31) == 0) red[tid >> 5] = loc;
    __syncthreads();
    if (tid == 0) {
        float t = 0.f;
        for (int i = 0; i < 8; ++i) t += red[i];
        invn = 1.0f / sqrtf(t);
    }
    __syncthreads();
    const float iv = invn;
    f[(size_t)b * Dv + tid]       = s0 * iv;
    f[(size_t)b * Dv + tid + 256] = s1 * iv;
}

// ---------------------------------------------------------------------------
// Kernel 3: per (c,m) softmax diagonal -> w[c,m] = exp(gamma*log2((1+eps)/(p_cc+eps)))
// ---------------------------------------------------------------------------
__global__ __launch_bounds__(256) void pcc_kernel(
    const float* __restrict__ keys_all, const float* __restrict__ tf,
    const int* __restrict__ indices, const float* __restrict__ gamma_p,
    float* __restrict__ wbuf)
{
    const int c    = blockIdx.x;
    const int tid  = threadIdx.x;
    const int lane = tid & 31;
    const int wave = tid >> 5;

    __shared__ int   sidx[NFv];
    __shared__ float Ttile[20 * NFv];       // 20 KB
    __shared__ float Lbuf[Mv * Cv];         // 25.6 KB

    sidx[tid] = indices[c * NFv + tid];
    __syncthreads();

    // cache gathered key rows for this wave's 8 m values
    float kreg[8][8];
#pragma unroll
    for (int mi = 0; mi < 8; ++mi) {
        const float* kr = keys_all + ((size_t)c * Mv + (wave * 8 + mi)) * Dv;
#pragma unroll
        for (int i = 0; i < 8; ++i) kreg[mi][i] = kr[sidx[lane + 32 * i]];
    }

    for (int jt = 0; jt < 5; ++jt) {
        for (int l = 0; l < 20; ++l)
            Ttile[l * NFv + tid] = tf[(size_t)(jt * 20 + l) * Dv + sidx[tid]];
        __syncthreads();

        for (int mi = 0; mi < 8; ++mi) {
            for (int l = 0; l < 20; ++l) {
                float s = 0.f;
#pragma unroll
                for (int i = 0; i < 8; ++i)
                    s += kreg[mi][i] * Ttile[l * NFv + lane + 32 * i];
#pragma unroll
                for (int off = 16; off > 0; off >>= 1) s += __shfl_xor(s, off, 32);
                if (lane == 0) Lbuf[(wave * 8 + mi) * Cv + jt * 20 + l] = s;
            }
        }
        __syncthreads();
    }

    const float gmm = *gamma_p;
    for (int mi = 0; mi < 8; ++mi) {
        const int m = wave * 8 + mi;
        const float* L = &Lbuf[m * Cv];
        float mx = -3.0e38f;
        for (int j = lane; j < Cv; j += 32) mx = fmaxf(mx, L[j]);
#pragma unroll
        for (int off = 16; off > 0; off >>= 1) mx = fmaxf(mx, __shfl_xor(mx, off, 32));
        float se = 0.f;
        for (int j = lane; j < Cv; j += 32) se += expf(L[j] - mx);
#pragma unroll
        for (int off = 16; off > 0; off >>= 1) se += __shfl_xor(se, off, 32);
        if (lane == 0) {
            const float p  = expf(L[c] - mx) / se;
            const float kl = log2f((1.0f + EPSv) / (p + EPSv));
            wbuf[c * Mv + m] = expf(kl * gmm);
        }
    }
}

// ---------------------------------------------------------------------------
// Kernel 4: kw[c,f] = sum_m keys_all[c,m,idx[c,f]] * w[c,m]
// ---------------------------------------------------------------------------
__global__ __launch_bounds__(256) void kw_kernel(
    const float* __restrict__ keys_all, const int* __restrict__ indices,
    const float* __restrict__ wbuf, float* __restrict__ kw)
{
    const int c  = blockIdx.x;
    const int fi = threadIdx.x;
    const int id = indices[c * NFv + fi];
    float s = 0.f;
    for (int m = 0; m < Mv; ++m)
        s += keys_all[((size_t)c * Mv + m) * Dv + id] * wbuf[c * Mv + m];
    kw[c * NFv + fi] = s;
}

// ---------------------------------------------------------------------------
// Kernel 5: out[b,c] = alpha*exp(beta*cache - beta) + exp(ls) * <f_b, text_c>
// ---------------------------------------------------------------------------
__global__ __launch_bounds__(256) void final_kernel(
    const float* __restrict__ f, const float* __restrict__ tf,
    const int* __restrict__ indices, const float* __restrict__ kw,
    const float* __restrict__ ls_p, const float* __restrict__ alpha_p,
    const float* __restrict__ beta_p, float* __restrict__ out)
{
    const int g = blockIdx.x * 256 + threadIdx.x;   // 25*256 == 6400 == Bv*Cv
    const int b = g / Cv;
    const int c = g % Cv;
    const float* frow = f + (size_t)b * Dv;

    float clip = 0.f;
    for (int d = 0; d < Dv; ++d) clip += frow[d] * tf[(size_t)c * Dv + d];

    float cache = 0.f;
    for (int t = 0; t < NFv; ++t)
        cache += frow[indices[c * NFv + t]] * kw[c * NFv + t];
    cache *= (1.0f / (float)Mv);

    const float beta = *beta_p;
    out[g] = (*alpha_p) * expf(beta * cache - beta) + expf(*ls_p) * clip;
}

// ---------------------------------------------------------------------------
extern "C" void kernel_launch(void* const* d_in, const int* in_sizes, int n_in,
                              void* d_out, int out_size, void* d_ws, size_t ws_size,
                              hipStream_t stream)
{
    const float* image   = (const float*)d_in[0];
    const float* Wenc    = (const float*)d_in[1];
    const float* tf      = (const float*)d_in[2];
    const float* keys    = (const float*)d_in[3];
    const float* ls      = (const float*)d_in[4];
    const int*   indices = (const int*)  d_in[5];
    const float* alpha   = (const float*)d_in[6];
    const float* beta    = (const float*)d_in[7];
    const float* gamma   = (const float*)d_in[8];
    float* out = (float*)d_out;

    float* ws      = (float*)d_ws;
    float* partial = ws;                                   // KSPLIT*Bv*Dv = 1048576 floats
    float* f       = partial + (size_t)KSPLIT * Bv * Dv;   // Bv*Dv
    float* wbuf    = f + (size_t)Bv * Dv;                  // Cv*Mv
    float* kw      = wbuf + Cv * Mv;                       // Cv*NFv

    gemm_partial_kernel<<<dim3(Dv / 64, KSPLIT), 256, 0, stream>>>(image, Wenc, partial);
    reduce_norm_kernel <<<Bv, 256, 0, stream>>>(partial, f);
    pcc_kernel         <<<Cv, 256, 0, stream>>>(keys, tf, indices, gamma, wbuf);
    kw_kernel          <<<Cv, 256, 0, stream>>>(keys, indices, wbuf, kw);
    final_kernel       <<<(Bv * Cv) / 256, 256, 0, stream>>>(f, tf, indices, kw, ls, alpha, beta, out);
}